// CrfLM_49280454754910
// MI455X (gfx1250) — compile-verified
//
#include <hip/hip_runtime.h>
#include <hip/hip_bf16.h>
#include <cstddef>

typedef __attribute__((ext_vector_type(2))) float v2f;
typedef __attribute__((ext_vector_type(8))) float v8f;

#define BATCH     32
#define SEQLEN    1024
#define DIM       1024
#define C         66        // NUM_TAGS
#define NPAD      80        // C padded to multiple of 16
#define START_IDX 64
#define STOP_IDX  65
#define IMPOSSIBLE (-10000.0f)

// ---------------------------------------------------------------------------
// Kernel 0: transpose + zero-pad fc_w [1024,66] -> Bt [80][1024]
// ---------------------------------------------------------------------------
__global__ void crf_transpose_w(const float* __restrict__ fc_w,
                                float* __restrict__ Bt) {
    int idx = blockIdx.x * blockDim.x + threadIdx.x;   // 80*1024 elems
    if (idx >= NPAD * DIM) return;
    int n = idx / DIM;
    int k = idx - n * DIM;
    Bt[idx] = (n < C) ? fc_w[k * C + n] : 0.0f;
}

// ---------------------------------------------------------------------------
// Kernel 1: feats = features @ fc_w + fc_b via v_wmma_f32_16x16x4_f32
// One wave per 16-row M strip; 5 N-tiles (80 cols) per wave; K=1024 in steps of 4.
// A frag layout: lane half h=lane/16, row m=lane%16, a[v]=A[m][k+2h+v]
// B frag layout: col n=lane%16,               b[v]=Bt[n][k+2h+v]
// D layout: d[r] = D[r+8h][lane%16]
// ---------------------------------------------------------------------------
__global__ __launch_bounds__(128) void crf_gemm_wmma(
    const float* __restrict__ A,      // [32768, 1024]
    const float* __restrict__ Bt,     // [80, 1024] (transposed, padded)
    const float* __restrict__ bias,   // [66]
    float* __restrict__ feats) {      // [32768, 66]
    const int wave = threadIdx.x >> 5;
    const int lane = threadIdx.x & 31;
    const int mtile = blockIdx.x * 4 + wave;          // 2048 tiles total
    const int m0 = mtile * 16;
    const int row = lane & 15;
    const int half = lane >> 4;

    const float2* a2 = (const float2*)(A + (size_t)(m0 + row) * DIM);

    v8f acc[5];
#pragma unroll
    for (int t = 0; t < 5; ++t) acc[t] = (v8f){0.f,0.f,0.f,0.f,0.f,0.f,0.f,0.f};

    for (int k = 0; k < DIM; k += 4) {
        const int kk = k + 2 * half;                  // even
        float2 av = a2[kk >> 1];                      // A[m][kk], A[m][kk+1]
        v2f afrag; afrag.x = av.x; afrag.y = av.y;
#pragma unroll
        for (int t = 0; t < 5; ++t) {
            const int n = t * 16 + row;
            float2 bv = ((const float2*)(Bt + (size_t)n * DIM))[kk >> 1];
            v2f bfrag; bfrag.x = bv.x; bfrag.y = bv.y;
            acc[t] = __builtin_amdgcn_wmma_f32_16x16x4_f32(
                false, afrag, false, bfrag, (short)0, acc[t], false, false);
        }
    }
    // store + bias (skip padded columns)
#pragma unroll
    for (int t = 0; t < 5; ++t) {
        const int n = t * 16 + row;
        if (n < C) {
            const float bv = bias[n];
#pragma unroll
            for (int r = 0; r < 8; ++r) {
                const int m = m0 + r + 8 * half;
                feats[(size_t)m * C + n] = acc[t][r] + bv;
            }
        }
    }
}

// ---------------------------------------------------------------------------
// Kernel 2: CRF forward partition function. One block (96 thr = 3 waves) per batch.
// Exp-domain scan: E=exp(T) in LDS; per step: M=max(s); p=exp(s-M);
// dot_j = E[j,:]·p ; s_j = (M + log dot_j + emit_j)*m + s_j*(1-m)
// ---------------------------------------------------------------------------
__global__ __launch_bounds__(96) void crf_forward_scan(
    const float* __restrict__ feats,   // [B, L, C]
    const float* __restrict__ masks,   // [B, L]
    const float* __restrict__ trans,   // [C, C]
    float* __restrict__ out) {         // out[0..31] forward scores
    const int tid  = threadIdx.x;
    const int lane = tid & 31;
    const int wave = tid >> 5;
    const int b    = blockIdx.x;

    __shared__ float E[C * C];     // exp(transitions), row-major [j][i]
    __shared__ float sS[96];       // scores, padded with -3e38
    __shared__ float pS[C + 2];    // exp(s - M)
    __shared__ float Msh;
    __shared__ float wsum[3];

    for (int idx = tid; idx < C * C; idx += 96) E[idx] = __expf(trans[idx]);
    sS[tid] = (tid == START_IDX) ? 0.0f : ((tid < C) ? IMPOSSIBLE : -3.0e38f);
    __syncthreads();

    const float* emitB = feats + (size_t)b * SEQLEN * C;
    const float* maskB = masks + (size_t)b * SEQLEN;

    for (int t = 0; t < SEQLEN; ++t) {
        // --- max over sS[0..95] by wave 0 ---
        if (wave == 0) {
            float m = fmaxf(fmaxf(sS[lane], sS[lane + 32]), sS[lane + 64]);
#pragma unroll
            for (int o = 16; o > 0; o >>= 1) m = fmaxf(m, __shfl_xor(m, o, 32));
            if (lane == 0) Msh = m;
        }
        __syncthreads();
        const float M = Msh;
        if (tid < C) pS[tid] = __expf(sS[tid] - M);
        __syncthreads();

        float snew = 0.0f;
        if (tid < C) {
            const float2* Er = (const float2*)(E + tid * C);   // tid*66*4 % 8 == 0
            const float2* pr = (const float2*)pS;
            float acc0 = 0.0f, acc1 = 0.0f;
#pragma unroll
            for (int i = 0; i < C / 2; ++i) {
                float2 e = Er[i];
                float2 p = pr[i];
                acc0 = fmaf(e.x, p.x, acc0);
                acc1 = fmaf(e.y, p.y, acc1);
            }
            const float dot = acc0 + acc1;
            float v = (dot > 0.0f) ? (M + __logf(dot)) : -1.0e30f;
            v += emitB[(size_t)t * C + tid];
            const float mt = maskB[t];
            snew = v * mt + sS[tid] * (1.0f - mt);
        }
        __syncthreads();
        if (tid < C) sS[tid] = snew;
    }
    __syncthreads();

    // --- final logsumexp over sS[j] + T[STOP, j] ---
    if (tid < C) sS[tid] += trans[STOP_IDX * C + tid];
    __syncthreads();
    if (wave == 0) {
        float m = fmaxf(fmaxf(sS[lane], sS[lane + 32]), sS[lane + 64]);
#pragma unroll
        for (int o = 16; o > 0; o >>= 1) m = fmaxf(m, __shfl_xor(m, o, 32));
        if (lane == 0) Msh = m;
    }
    __syncthreads();
    const float M = Msh;
    float e = (tid < C) ? __expf(sS[tid] - M) : 0.0f;
#pragma unroll
    for (int o = 16; o > 0; o >>= 1) e += __shfl_xor(e, o, 32);
    if (lane == 0) wsum[wave] = e;
    __syncthreads();
    if (tid == 0) out[b] = M + __logf(wsum[0] + wsum[1] + wsum[2]);
}

// ---------------------------------------------------------------------------
// Kernel 3: gold path score. One wave per batch.
// ---------------------------------------------------------------------------
__global__ __launch_bounds__(32) void crf_gold_score(
    const float* __restrict__ feats,   // [B, L, C]
    const int*   __restrict__ ys,      // [B, L]
    const float* __restrict__ masks,   // [B, L]
    const float* __restrict__ trans,   // [C, C]
    float* __restrict__ out) {         // out[32..63]
    const int b = blockIdx.x;
    const int lane = threadIdx.x;
    const int* yB = ys + (size_t)b * SEQLEN;
    const float* mB = masks + (size_t)b * SEQLEN;
    const float* fB = feats + (size_t)b * SEQLEN * C;

    float sum = 0.0f, len = 0.0f;
    for (int t = lane; t < SEQLEN; t += 32) {
        const float m = mB[t];
        const int tag  = yB[t];
        const int prev = (t == 0) ? START_IDX : yB[t - 1];
        const float emit = fB[(size_t)t * C + tag];
        const float tr   = trans[tag * C + prev];
        sum = fmaf(tr + emit, m, sum);
        len += m;
    }
#pragma unroll
    for (int o = 16; o > 0; o >>= 1) {
        sum += __shfl_xor(sum, o, 32);
        len += __shfl_xor(len, o, 32);
    }
    if (lane == 0) {
        const int L = (int)(len + 0.5f);
        const int last = (L == 0) ? START_IDX : yB[L - 1];
        out[BATCH + b] = sum + trans[STOP_IDX * C + last];
    }
}

// ---------------------------------------------------------------------------
extern "C" void kernel_launch(void* const* d_in, const int* in_sizes, int n_in,
                              void* d_out, int out_size, void* d_ws, size_t ws_size,
                              hipStream_t stream) {
    const float* features = (const float*)d_in[0];   // [32,1024,1024]
    const int*   ys       = (const int*)  d_in[1];   // [32,1024]
    const float* masks    = (const float*)d_in[2];   // [32,1024]
    const float* fc_w     = (const float*)d_in[3];   // [1024,66]
    const float* fc_b     = (const float*)d_in[4];   // [66]
    const float* trans    = (const float*)d_in[5];   // [66,66]
    float* out = (float*)d_out;                      // [64]

    float* Bt    = (float*)d_ws;                     // [80,1024]
    float* feats = Bt + (size_t)NPAD * DIM;          // [32768, 66]

    // 0) transpose+pad fc_w
    crf_transpose_w<<<(NPAD * DIM + 255) / 256, 256, 0, stream>>>(fc_w, Bt);
    // 1) projection GEMM via f32 WMMA: 2048 M-tiles, 4 waves/block
    crf_gemm_wmma<<<512, 128, 0, stream>>>(features, Bt, fc_b, feats);
    // 2) CRF forward scan, one WGP per batch element
    crf_forward_scan<<<BATCH, 96, 0, stream>>>(feats, masks, trans, out);
    // 3) gold path score
    crf_gold_score<<<BATCH, 32, 0, stream>>>(feats, ys, masks, trans, out);
}